// QLSTM_65481071399708
// MI455X (gfx1250) — compile-verified
//
#include <hip/hip_runtime.h>

typedef __attribute__((ext_vector_type(2))) float v2f;
typedef __attribute__((ext_vector_type(8))) float v8f;

namespace {

constexpr int kT = 32;        // timesteps
constexpr int kB = 128;       // batch
constexpr int kD = 64;        // input dim
constexpr int kN = 8;         // qubits
constexpr int kK = 72;        // contraction dim (D + N)
constexpr int SPB = 4;        // samples per block
constexpr int THREADS = SPB * 4 * 32;  // 512: 4 samples x 4 gate types x wave32
constexpr int AROW = 72;      // A-operand row length in LDS

__device__ __forceinline__ float sigm(float x) { return 1.0f / (1.0f + __expf(-x)); }

// Plain RY on a cross-lane bit D (first wire only; later wires are fused with CNOT).
template <int D>
__device__ __forceinline__ void ry_lane(float (&u)[8], float (&v)[8], float c, float s, int lane) {
  const float sg = (lane & D) ? s : -s;
#pragma unroll
  for (int r = 0; r < 8; ++r) {
    float pu = __shfl_xor(u[r], D, 32);
    float pv = __shfl_xor(v[r], D, 32);
    u[r] = c * u[r] + sg * pu;
    v[r] = c * v[r] + sg * pv;
  }
}

// Fused CNOT(ctrl = lane bit DC, tgt = lane bit D) immediately followed by RY on wire D.
// ctrl==1 swaps the pair before the rotation; realized as selects on the shuffled pair,
// so the whole CNOT+RY pair costs a single shuffle per value.
template <int DC, int D>
__device__ __forceinline__ void cnot_ry_lane(float (&u)[8], float (&v)[8], float c, float s,
                                             int lane) {
  const bool ctl = (lane & DC) != 0;
  const float sg = (lane & D) ? s : -s;
#pragma unroll
  for (int r = 0; r < 8; ++r) {
    float pu = __shfl_xor(u[r], D, 32);
    float pv = __shfl_xor(v[r], D, 32);
    float e0u = ctl ? pu : u[r], e1u = ctl ? u[r] : pu;
    float e0v = ctl ? pv : v[r], e1v = ctl ? v[r] : pv;
    u[r] = c * e0u + sg * e1u;
    v[r] = c * e0v + sg * e1v;
  }
}

// RY on an in-thread register bit (pairs r <-> r|D): pure VALU, no shuffles.
template <int D>
__device__ __forceinline__ void ry_reg(float (&u)[8], float (&v)[8], float c, float s) {
#pragma unroll
  for (int r = 0; r < 8; ++r) {
    if (!(r & D)) {
      float a0u = u[r], a1u = u[r | D];
      float a0v = v[r], a1v = v[r | D];
      u[r]     = c * a0u - s * a1u;
      u[r | D] = s * a0u + c * a1u;
      v[r]     = c * a0v - s * a1v;
      v[r | D] = s * a0v + c * a1v;
    }
  }
}

// Branch-free CNOT: ctrl = lane bit DC, tgt = register bit DT (cndmask pair swap).
template <int DC, int DT>
__device__ __forceinline__ void cnot_lane_reg(float (&u)[8], float (&v)[8], int lane) {
  const bool ctl = (lane & DC) != 0;
#pragma unroll
  for (int r = 0; r < 8; ++r) {
    if (!(r & DT)) {
      float a = u[r], b = u[r | DT];
      u[r] = ctl ? b : a;  u[r | DT] = ctl ? a : b;
      a = v[r];  b = v[r | DT];
      v[r] = ctl ? b : a;  v[r | DT] = ctl ? a : b;
    }
  }
}

// CNOT with both ctrl and tgt on register bits: compile-time register permutation (free).
template <int CB, int DT>
__device__ __forceinline__ void cnot_reg_reg(float (&u)[8], float (&v)[8]) {
#pragma unroll
  for (int r = 0; r < 8; ++r) {
    if ((r & CB) && !(r & DT)) {
      float t = u[r]; u[r] = u[r | DT]; u[r | DT] = t;
      t = v[r]; v[r] = v[r | DT]; v[r | DT] = t;
    }
  }
}

}  // namespace

__global__ __launch_bounds__(THREADS) void qlstm_fused(
    const float* __restrict__ xin,   // [32,128,64]
    const float* __restrict__ W,     // [16,72] row-major: W[n][k]
    const float* __restrict__ bvec,  // [16]
    const float* __restrict__ pf, const float* __restrict__ pin,
    const float* __restrict__ pupd, const float* __restrict__ po,
    float* __restrict__ out)         // outs[32*128*8] ++ hx[128*8] ++ cx[128*8]
{
  // A operand, double buffered and padded to all 16 WMMA rows.
  // Row m: cols 0..63 = x[t, sample m], cols 64..71 = h[sample m]. Rows >= SPB stay zero.
  __shared__ __align__(16) float a_lds[2][16][AROW];
  __shared__ __align__(16) float w_lds[16][AROW];  // B operand: w_lds[n][k] = W[n][k]
  __shared__ float qv_lds[SPB][kN];                // only first 8 projection outputs are used
  __shared__ float z_lds[SPB][4][kN];
  __shared__ float c_lds[SPB][kN];

  const int tid  = threadIdx.x;
  const int lane = tid & 31;
  const int wid  = tid >> 5;   // 0..15
  const int s    = wid >> 2;   // sample-in-block
  const int g    = wid & 3;    // gate type: 0=f 1=i 2=g 3=o
  const int bid  = blockIdx.x; // 0..31

  // Per-wave RY constants (t-invariant)
  const float* prm = (g == 0) ? pf : (g == 1) ? pin : (g == 2) ? pupd : po;
  float cw[8], sw[8];
#pragma unroll
  for (int w = 0; w < 8; ++w) __sincosf(prm[w] * 0.5f, &sw[w], &cw[w]);

  // ---- one-time init ----
  for (int i = tid; i < 2 * 16 * AROW; i += THREADS) ((float*)a_lds)[i] = 0.0f;
  for (int i = tid; i < 16 * kK; i += THREADS) ((float*)w_lds)[i] = W[i];
  if (tid < SPB * kN) c_lds[tid >> 3][tid & 7] = 0.0f;
  const float bias = (lane < kN) ? bvec[lane] : 0.0f;
  __syncthreads();
  // stage x(0) into buffer 0 (after the zero-fill barrier)
  if (tid < SPB * 16) {
    int m = tid >> 4, d4 = tid & 15;
    float4 x0 = ((const float4*)xin)[(((0 * kB) + bid * SPB + m) * kD >> 2) + d4];
    *(float4*)&a_lds[0][m][d4 * 4] = x0;
  }
  __syncthreads();

  for (int t = 0; t < kT; ++t) {
    const int cb = t & 1, nb = cb ^ 1;

    // ---- prefetch x(t+1) into registers; committed to LDS at end of step ----
    float4 xx = {0.f, 0.f, 0.f, 0.f};
    const int tn = (t + 1 < kT) ? (t + 1) : (kT - 1);
    if (tid < SPB * 16) {
      int m = tid >> 4, d4 = tid & 15;
      xx = ((const float4*)xin)[(((tn * kB) + bid * SPB + m) * kD >> 2) + d4];
    }

    // ---- projection: qvec = concat(x,h) @ W.T + b via V_WMMA_F32_16X16X4_F32 ----
    if (wid == 0) {
      const int mn   = lane & 15;        // A row M / B col N per ISA 16x4 fragment layout
      const int koff = (lane >> 4) * 2;  // lanes 16..31 carry K+2/K+3
      const float* Ab = &a_lds[cb][mn][koff];
      const float* Bb = &w_lds[mn][koff];
      v8f C = {0.f, 0.f, 0.f, 0.f, 0.f, 0.f, 0.f, 0.f};
#pragma unroll
      for (int kc = 0; kc < 18; ++kc) {
        v2f A  = {Ab[kc * 4], Ab[kc * 4 + 1]};   // ds_load_b64, immediate offset
        v2f Bv = {Bb[kc * 4], Bb[kc * 4 + 1]};
        C = __builtin_amdgcn_wmma_f32_16x16x4_f32(false, A, false, Bv, (short)0, C,
                                                  false, false);
      }
      // D layout: VGPR r, lanes 0..15 -> (M=r, N=lane); circuits use only N<8
      if (lane < kN) {
#pragma unroll
        for (int r = 0; r < SPB; ++r) qv_lds[r][lane] = C[r] + bias;
      }
    }
    __syncthreads();

    // ---- quantum circuit: wave (s,g); lane holds amplitudes k = lane*8 + r ----
    // wire w <-> bit (7-w) of k: wires 0..4 = lane bits 4..0, wires 5..7 = reg bits 2..0
    float cx_[8], sx_[8];
#pragma unroll
    for (int w = 0; w < 8; ++w) __sincosf(qv_lds[s][w] * 0.5f, &sx_[w], &cx_[w]);

    // product state after RX layer: amp = (-i)^pop(k) * prod(bit ? sin : cos)
    float rl = 1.0f;
#pragma unroll
    for (int w = 0; w < 5; ++w) rl *= ((lane >> (4 - w)) & 1) ? sx_[w] : cx_[w];
    const int popl = __popc(lane & 31);

    float u[8], v[8];
#pragma unroll
    for (int r = 0; r < 8; ++r) {
      float rr = ((r & 4) ? sx_[5] : cx_[5]) * ((r & 2) ? sx_[6] : cx_[6]) *
                 ((r & 1) ? sx_[7] : cx_[7]);
      float amp = rl * rr;
      int p = (popl + __popc(r)) & 3;  // phase (-i)^p
      u[r] = (p == 0) ? amp : ((p == 2) ? -amp : 0.0f);
      v[r] = (p == 3) ? amp : ((p == 1) ? -amp : 0.0f);
    }

    // RY(w) then CNOT(w, w+1); cross-lane CNOTs fused into the next RY's butterfly.
    ry_lane<16>(u, v, cw[0], sw[0], lane);            // RY0
    cnot_ry_lane<16, 8>(u, v, cw[1], sw[1], lane);    // CNOT(0,1) + RY1
    cnot_ry_lane<8, 4>(u, v, cw[2], sw[2], lane);     // CNOT(1,2) + RY2
    cnot_ry_lane<4, 2>(u, v, cw[3], sw[3], lane);     // CNOT(2,3) + RY3
    cnot_ry_lane<2, 1>(u, v, cw[4], sw[4], lane);     // CNOT(3,4) + RY4
    cnot_lane_reg<1, 4>(u, v, lane);                  // CNOT(4,5)
    ry_reg<4>(u, v, cw[5], sw[5]);                    // RY5
    cnot_reg_reg<4, 2>(u, v);                         // CNOT(5,6)  (free)
    ry_reg<2>(u, v, cw[6], sw[6]);                    // RY6
    cnot_reg_reg<2, 1>(u, v);                         // CNOT(6,7)  (free)
    ry_reg<1>(u, v, cw[7], sw[7]);                    // RY7
    // CNOT(7,0): ctrl = reg bit 0, tgt = lane bit 16 -> only odd registers exchange
#pragma unroll
    for (int r = 1; r < 8; r += 2) {
      u[r] = __shfl_xor(u[r], 16, 32);
      v[r] = __shfl_xor(v[r], 16, 32);
    }

    // ---- measure <Z_w>. Wires 0..4 are single-bit sign masks over lanes:
    // one Walsh-Hadamard butterfly of psum yields all of them (lane m holds coeff F(m)).
    float psum = 0.0f, z5 = 0.0f, z6 = 0.0f, z7 = 0.0f;
#pragma unroll
    for (int r = 0; r < 8; ++r) {
      float p = u[r] * u[r] + v[r] * v[r];
      psum += p;
      z5 += (r & 4) ? -p : p;
      z6 += (r & 2) ? -p : p;
      z7 += (r & 1) ? -p : p;
    }
    float zw = psum;
#pragma unroll
    for (int dd = 1; dd <= 16; dd <<= 1) {
      float pw = __shfl_xor(zw, dd, 32);
      zw = (lane & dd) ? (pw - zw) : (zw + pw);
      z5 += __shfl_xor(z5, dd, 32);
      z6 += __shfl_xor(z6, dd, 32);
      z7 += __shfl_xor(z7, dd, 32);
    }
    // wire w (0..4) = coefficient at lane (1 << (4-w))
    if (lane && !(lane & (lane - 1))) z_lds[s][g][4 - (__ffs(lane) - 1)] = zw;
    if (lane == 0) { z_lds[s][g][5] = z5; z_lds[s][g][6] = z6; z_lds[s][g][7] = z7; }
    __syncthreads();

    // ---- LSTM cell update; h(t+1) goes into the *next* A buffer ----
    if (tid < SPB * kN) {
      int ss = tid >> 3, j = tid & 7;
      float fg = sigm(z_lds[ss][0][j]);
      float ig = sigm(z_lds[ss][1][j]);
      float gg = tanhf(z_lds[ss][2][j]);
      float og = sigm(z_lds[ss][3][j]);
      float cc = fg * c_lds[ss][j] + ig * gg;
      c_lds[ss][j] = cc;
      float hh = og * tanhf(cc);
      a_lds[nb][ss][kD + j] = hh;
      out[((t * kB) + (bid * SPB + ss)) * kN + j] = hh;
    }
    // commit the prefetched x(t+1) tile into the next A buffer
    if (tid < SPB * 16) {
      int m = tid >> 4, d4 = tid & 15;
      *(float4*)&a_lds[nb][m][d4 * 4] = xx;
    }
    __syncthreads();
  }

  // ---- final hx, cx (h after t=31 landed in buffer (kT & 1) == 0) ----
  if (tid < SPB * kN) {
    int ss = tid >> 3, j = tid & 7;
    int b = bid * SPB + ss;
    out[kT * kB * kN + b * kN + j] = a_lds[0][ss][kD + j];
    out[kT * kB * kN + kB * kN + b * kN + j] = c_lds[ss][j];
  }
}

extern "C" void kernel_launch(void* const* d_in, const int* in_sizes, int n_in,
                              void* d_out, int out_size, void* d_ws, size_t ws_size,
                              hipStream_t stream) {
  (void)in_sizes; (void)n_in; (void)out_size; (void)d_ws; (void)ws_size;
  const float* xin  = (const float*)d_in[0];
  const float* W    = (const float*)d_in[1];
  const float* bvec = (const float*)d_in[2];
  const float* pf   = (const float*)d_in[3];
  const float* pin  = (const float*)d_in[4];
  const float* pupd = (const float*)d_in[5];
  const float* po   = (const float*)d_in[6];
  qlstm_fused<<<kB / SPB, THREADS, 0, stream>>>(xin, W, bvec, pf, pin, pupd, po,
                                                (float*)d_out);
}